// SpatialEmbedding_64604898066679
// MI455X (gfx1250) — compile-verified
//
#include <hip/hip_runtime.h>
#include <hip/hip_bf16.h>

// SpatialEmbedding: out[b,c,i,j] = x[b,c,i,j] + emb[i/4, j/4, c]
// x: [16,384,64,64] f32, emb: [1,16,16,384] f32.
// Pure streaming add: ~192MB traffic -> HBM-bound (~8.3us @ 23.3TB/s).
// One block per (b,c) plane; per-channel 256-float emb slice staged in LDS
// via gfx1250 async global->LDS copy.

typedef float v4f __attribute__((ext_vector_type(4)));

#define NCH   384
#define PLANE 4096   // 64*64 floats per (b,c) plane
#define NPLANES (16 * NCH)

#if defined(__AMDGCN__)
typedef __attribute__((address_space(1))) int gint_t;   // global (AS1)
typedef __attribute__((address_space(3))) int lint_t;   // LDS    (AS3)
#endif

__global__ __launch_bounds__(256) void spatial_emb_add_kernel(
    const float* __restrict__ x,
    const float* __restrict__ emb,
    float* __restrict__ out)
{
    __shared__ float se[256];   // emb[cell, c] for this block's channel, cell=ti*16+tj

    const int t = threadIdx.x;
    const int p = blockIdx.x;        // plane index = b*384 + c
    const int c = p % NCH;

    // ---- Stage the 256-entry per-channel embedding slice into LDS ----
    // se[cell] = emb[cell*384 + c]  (strided gather, one b32 per lane)
    const float* gsrc = emb + (size_t)t * NCH + c;
#if defined(__gfx1250__) && __has_builtin(__builtin_amdgcn_global_load_async_to_lds_b32)
    __builtin_amdgcn_global_load_async_to_lds_b32(
        (gint_t*)gsrc, (lint_t*)&se[t], /*offset=*/0, /*cpol=*/0);
  #if __has_builtin(__builtin_amdgcn_s_wait_asynccnt)
    __builtin_amdgcn_s_wait_asynccnt(0);
  #else
    asm volatile("s_wait_asynccnt 0x0" ::: "memory");
  #endif
#else
    se[t] = __ldg(gsrc);
#endif
    __syncthreads();

    // ---- Stream one 16KB plane: 1024 float4 chunks, 256 threads x 4 ----
    const v4f* __restrict__ xb = (const v4f*)(x + (size_t)p * PLANE);
    v4f* __restrict__ ob       = (v4f*)(out + (size_t)p * PLANE);

#pragma unroll
    for (int k = 0; k < 4; ++k) {
        const int q  = t + 256 * k;      // float4 index within plane
        const int i  = q >> 4;           // pixel row (0..63)
        const int j4 = q & 15;           // float4 column == j/4 (aligned)
        const float e = se[((i >> 2) << 4) + j4];

        // keep the load pipe ahead (lowered to global_prefetch_b8 on gfx1250;
        // speculative, OOB past buffer end is dropped by HW)
        __builtin_prefetch((const void*)(xb + q + 256), 0, 1);

        v4f v = xb[q];
        v.x += e; v.y += e; v.z += e; v.w += e;

        // write-once output: non-temporal so it doesn't evict x from L2
        __builtin_nontemporal_store(v, ob + q);
    }
}

extern "C" void kernel_launch(void* const* d_in, const int* in_sizes, int n_in,
                              void* d_out, int out_size, void* d_ws, size_t ws_size,
                              hipStream_t stream) {
    const float* x   = (const float*)d_in[0];   // [16,384,64,64]
    const float* emb = (const float*)d_in[1];   // [1,16,16,384]
    // d_in[2] is grid_size (==16), baked into the index math above.
    float* out = (float*)d_out;

    spatial_emb_add_kernel<<<NPLANES, 256, 0, stream>>>(x, emb, out);
}